// GINNet_64295660421276
// MI455X (gfx1250) — compile-verified
//
#include <hip/hip_runtime.h>

typedef __attribute__((ext_vector_type(2))) float v2f;
typedef __attribute__((ext_vector_type(8))) float v8f;

#define NN 50000
#define CH 128

// ---------------- zero workspace (float4 vectorized) ----------------
__global__ void zero_f4(float4* __restrict__ p, int n4) {
  int i = blockIdx.x * blockDim.x + threadIdx.x;
  if (i < n4) p[i] = make_float4(0.f, 0.f, 0.f, 0.f);
}

// ---------------- edge scatter-add: agg[dst] += X[src] ----------------
// one wave (32 lanes) per edge; each lane moves 4 consecutive floats.
// accumulator is L2-resident (25.6 MB << 192 MB L2), hardware f32 atomics.
__global__ void scatter_add(const float* __restrict__ X,
                            const int* __restrict__ src,
                            const int* __restrict__ dst,
                            float* __restrict__ agg, int E) {
  long long t = (long long)blockIdx.x * blockDim.x + threadIdx.x;
  int e = (int)(t >> 5);
  if (e >= E) return;
  int lane = (int)(t & 31);
  int s = src[e], d = dst[e];
  float4 v = *((const float4*)(X + (size_t)s * CH) + lane);
  float* a = agg + (size_t)d * CH + lane * 4;
  unsafeAtomicAdd(a + 0, v.x);   // global_atomic_add_f32
  unsafeAtomicAdd(a + 1, v.y);
  unsafeAtomicAdd(a + 2, v.z);
  unsafeAtomicAdd(a + 3, v.w);
}

// ---------------- fused (A [+ AGG]) @ W + bias [, ReLU] via FP32 WMMA ----------------
// Each wave owns one 16-wide n-tile and MT_PER_WAVE m-tiles. The whole B panel
// for its n-tile (K/4 x v2f = 64 VGPRs for K=128) is preloaded into registers
// and reused across all m-tiles, so the steady-state inner loop is only
// A-pair loads feeding the V_WMMA_F32_16X16X4_F32 accumulation chain.
template <bool ADD_AGG, bool RELU, int K, int MT_PER_WAVE>
__global__ void gemm_relu_wmma(const float* __restrict__ A,
                               const float* __restrict__ AGG,
                               const float* __restrict__ W,     // K x N row-major
                               const float* __restrict__ bias,  // N
                               float* __restrict__ out,         // M x N
                               int M, int N) {
  const int lane = threadIdx.x & 31;
  const int wave = threadIdx.x >> 5;
  const int tilesN = N >> 4;
  const int mchunks = (M >> 4) / MT_PER_WAVE;
  const int work = blockIdx.x * (blockDim.x >> 5) + wave;
  const int chunk = work / tilesN;
  if (chunk >= mchunks) return;       // whole wave exits together -> EXEC all-ones at WMMA
  const int nt = work - chunk * tilesN;
  const int hi = lane >> 4;           // 0: lanes 0-15, 1: lanes 16-31
  const int lo = lane & 15;
  const int col = (nt << 4) + lo;

  // Preload all B fragments for this n-tile.
  // B 4x16 layout: VGPR0/1 hold rows K = k0 + hi*2 (+1) at col = n0 + (lane&15).
  v2f bf[K / 4];
#pragma unroll
  for (int i = 0; i < K / 4; ++i) {
    const int ka = 4 * i + (hi << 1);
    bf[i].x = W[(size_t)ka * N + col];
    bf[i].y = W[(size_t)(ka + 1) * N + col];
  }
  const float bv = bias[col];

  const int mt0 = chunk * MT_PER_WAVE;
  for (int mi = 0; mi < MT_PER_WAVE; ++mi) {
    const int m0 = (mt0 + mi) << 4;
    // A 16x4 layout: lane holds row m0+lo, K pair starting at k0 + hi*2.
    const float* arow = A + (size_t)(m0 + lo) * K + (hi << 1);
    const float* grow = ADD_AGG ? (AGG + (size_t)(m0 + lo) * K + (hi << 1)) : nullptr;

    v8f c;
#pragma unroll
    for (int r = 0; r < 8; ++r) c[r] = bv;

#pragma unroll
    for (int i = 0; i < K / 4; ++i) {
      v2f a;
      a.x = arow[4 * i];
      a.y = arow[4 * i + 1];
      if (ADD_AGG) {
        a.x += grow[4 * i];
        a.y += grow[4 * i + 1];
      }
      c = __builtin_amdgcn_wmma_f32_16x16x4_f32(
          /*neg_a=*/false, a, /*neg_b=*/false, bf[i],
          /*c_mod=*/(short)0, c, /*reuse_a=*/false, /*reuse_b=*/false);
    }

    // C/D layout: VGPR r -> row m0 + r + hi*8, col = n0 + (lane&15)
    float* orow = out + (size_t)(m0 + hi * 8) * N + col;
#pragma unroll
    for (int r = 0; r < 8; ++r) {
      float v = c[r];
      if (RELU) v = fmaxf(v, 0.f);
      orow[(size_t)r * N] = v;
    }
  }
}

// ---------------- head: out[n] = dot(H[n,0:64], w4) + b4 ----------------
// one wave per node; wave32 shuffle reduction.
__global__ void head_gemv(const float* __restrict__ H,
                          const float* __restrict__ w4,
                          const float* __restrict__ b4,
                          float* __restrict__ out, int M) {
  long long t = (long long)blockIdx.x * blockDim.x + threadIdx.x;
  int n = (int)(t >> 5);
  if (n >= M) return;
  int lane = (int)(t & 31);
  const float* h = H + (size_t)n * 64;
  float s = h[lane] * w4[lane] + h[lane + 32] * w4[lane + 32];
#pragma unroll
  for (int off = 16; off > 0; off >>= 1) s += __shfl_xor(s, off, 32);
  if (lane == 0) out[n] = s + b4[0];
}

extern "C" void kernel_launch(void* const* d_in, const int* in_sizes, int n_in,
                              void* d_out, int out_size, void* d_ws, size_t ws_size,
                              hipStream_t stream) {
  const float* x   = (const float*)d_in[0];
  const int*   ei  = (const int*)d_in[1];
  const int    E   = in_sizes[1] / 2;
  const int* src = ei;          // edge_index[0]
  const int* dst = ei + E;      // edge_index[1]
  const float* w1a = (const float*)d_in[2];
  const float* b1a = (const float*)d_in[3];
  const float* w1b = (const float*)d_in[4];
  const float* b1b = (const float*)d_in[5];
  const float* w2a = (const float*)d_in[6];
  const float* b2a = (const float*)d_in[7];
  const float* w2b = (const float*)d_in[8];
  const float* b2b = (const float*)d_in[9];
  const float* w3  = (const float*)d_in[10];
  const float* b3  = (const float*)d_in[11];
  const float* w4  = (const float*)d_in[12];
  const float* b4  = (const float*)d_in[13];
  float* outp = (float*)d_out;

  // workspace carve (agg reused as head buffer h3 once agg is dead)
  float* agg  = (float*)d_ws;                       // NN*CH
  float* bufA = agg  + (size_t)NN * CH;             // NN*CH
  float* bufB = bufA + (size_t)NN * CH;             // NN*CH
  float* h3   = agg;                                // NN*64 (aliases agg, safe)

  const int n4 = NN * CH / 4;
  dim3 zb(256), zg((n4 + 255) / 256);
  long long sthreads = (long long)E * 32;
  dim3 sb(256), sg((unsigned)((sthreads + 255) / 256));

  // 3125 m-tiles = 625 chunks of 5; waves = chunks * (N/16); 4 waves / block.
  constexpr int MTW = 5;
  auto gblocks = [](int M, int N) {
    int waves = ((M >> 4) / MTW) * (N >> 4);
    return (unsigned)((waves + 3) / 4);
  };

  // ---- conv1 ----
  zero_f4<<<zg, zb, 0, stream>>>((float4*)agg, n4);
  scatter_add<<<sg, sb, 0, stream>>>(x, src, dst, agg, E);
  gemm_relu_wmma<true,  true, CH, MTW><<<gblocks(NN, 128), 128, 0, stream>>>(x,    agg,     w1a, b1a, bufA, NN, 128);
  gemm_relu_wmma<false, true, CH, MTW><<<gblocks(NN, 128), 128, 0, stream>>>(bufA, nullptr, w1b, b1b, bufB, NN, 128);

  // ---- conv2 ----
  zero_f4<<<zg, zb, 0, stream>>>((float4*)agg, n4);
  scatter_add<<<sg, sb, 0, stream>>>(bufB, src, dst, agg, E);
  gemm_relu_wmma<true,  true, CH, MTW><<<gblocks(NN, 128), 128, 0, stream>>>(bufB, agg,     w2a, b2a, bufA, NN, 128);
  gemm_relu_wmma<false, true, CH, MTW><<<gblocks(NN, 128), 128, 0, stream>>>(bufA, nullptr, w2b, b2b, bufB, NN, 128);

  // ---- head ----
  gemm_relu_wmma<false, true, CH, MTW><<<gblocks(NN, 64), 128, 0, stream>>>(bufB, nullptr, w3, b3, h3, NN, 64);
  head_gemv<<<(NN * 32 + 255) / 256, 256, 0, stream>>>(h3, w4, b4, outp, NN);
}